// AddShift_mp_module_1d_37830071943707
// MI455X (gfx1250) — compile-verified
//
#include <hip/hip_runtime.h>
#include <stdint.h>

// AddShift: 3x (per-channel zero-filled shift + group/nk reduction).
// Memory-bound: ~260 MB total HBM traffic -> ~11 us floor at 23.3 TB/s.
// Stage each channel row once into LDS via CDNA5 async b128 global->LDS DMA,
// serve all three shift patterns from LDS; zero-fill handled by halo zeroing.

namespace {
constexpr int kNK      = 7;
constexpr int kCout    = 64;
constexpr int kGroups  = 4;
constexpr int kCin     = kGroups * kCout * kNK;   // 1792
constexpr int kB       = 8;
constexpr int kL       = 4096;
constexpr int kTile    = 512;
constexpr int kHalo    = 16;                      // staged halo; > max|shift|=15; 16B-chunk aligned
constexpr int kRow     = kTile + 2 * kHalo;       // 544 floats per staged row (2176 B = 136 chunks)
constexpr int kChunks  = kRow / 4;                // 136 x 16B chunks per row
constexpr int kNch     = kGroups * kNK;           // 28 channels per c_out
constexpr int kThreads = 256;
constexpr int kNtiles  = kL / kTile;              // 8
}

__global__ __launch_bounds__(kThreads) void addshift3_kernel(
    const float* __restrict__ x,
    const int*   __restrict__ idx1,
    const int*   __restrict__ idx2,
    const int*   __restrict__ idx3,
    float*       __restrict__ out)
{
  __shared__ __align__(16) float tile_lds[kNch * kRow];   // ~59.5 KB
  __shared__ int shift_lds[3][kNch];

  const int tileId = blockIdx.x % kNtiles;
  const int bc     = blockIdx.x / kNtiles;
  const int co     = bc % kCout;
  const int b      = bc / kCout;
  const int t0     = tileId * kTile;
  const int tid    = threadIdx.x;

  // Per-block shift table: s = 15 - 5*(idx[c] % 7), c = g*448 + co*7 + k.
  if (tid < 3 * kNch) {
    const int br = tid / kNch;
    const int j  = tid % kNch;
    const int g  = j / kNK;
    const int k  = j % kNK;
    const int cg = g * (kCout * kNK) + co * kNK + k;
    const int* idx = (br == 0) ? idx1 : (br == 1) ? idx2 : idx3;
    shift_lds[br][j] = 15 - 5 * (idx[cg] % kNK);
  }

  // ---- Stage 28 rows (with 16-elem halo) via async b128 global->LDS DMA. ----
  // Chunk validity is uniform per 16B chunk (t0 % 4 == 0, L % 4 == 0), so
  // out-of-range chunks are clamped (their LDS slots are zeroed below).
  const uint64_t xbase   = (uint64_t)(uintptr_t)x;
  const uint32_t ldsbase = (uint32_t)(uintptr_t)&tile_lds[0];
  for (int e = tid; e < kNch * kChunks; e += kThreads) {
    const int j  = e / kChunks;            // channel slot 0..27
    const int c4 = e - j * kChunks;        // chunk within row
    const int g  = j / kNK;
    const int k  = j - g * kNK;
    const int cg = g * (kCout * kNK) + co * kNK + k;
    int tp = t0 - kHalo + 4 * c4;          // chunk start t (multiple of 4)
    tp = ((unsigned)tp < (unsigned)kL) ? tp : 0;   // whole-chunk clamp
    const uint32_t goff  = (uint32_t)(((b * kCin + cg) * kL + tp) * 4);
    const uint32_t laddr = ldsbase + (uint32_t)(e * 16);
    asm volatile("global_load_async_to_lds_b128 %0, %1, %2"
                 :: "v"(laddr), "v"(goff), "s"(xbase)
                 : "memory");
  }
  asm volatile("s_wait_asynccnt 0" ::: "memory");
  __syncthreads();

  // ---- Zero halo slots whose source t is outside [0, L). Edge tiles only. ----
  if (t0 == 0) {
    for (int e = tid; e < kNch * kHalo; e += kThreads) {
      const int j = e >> 4, p = e & (kHalo - 1);
      tile_lds[j * kRow + p] = 0.f;
    }
  } else if (t0 == kL - kTile) {
    for (int e = tid; e < kNch * kHalo; e += kThreads) {
      const int j = e >> 4, p = e & (kHalo - 1);
      tile_lds[j * kRow + (kRow - kHalo) + p] = 0.f;
    }
  }
  if (t0 == 0 || t0 == kL - kTile) __syncthreads();

  // ---- Accumulate: both t positions per thread inside the channel loop. ----
  const int tl = tid;                      // positions t0+tl and t0+tl+256
  float a00 = 0.f, a01 = 0.f, a02 = 0.f;   // branch 1/2/3 at tl
  float a10 = 0.f, a11 = 0.f, a12 = 0.f;   // branch 1/2/3 at tl+256
  #pragma unroll
  for (int j = 0; j < kNch; ++j) {
    const int s0 = shift_lds[0][j];
    const int s1 = shift_lds[1][j];
    const int s2 = shift_lds[2][j];
    const float* rowp = &tile_lds[j * kRow + kHalo + tl];
    a00 += rowp[-s0];
    a10 += rowp[kThreads - s0];
    a01 += rowp[-s1];
    a11 += rowp[kThreads - s1];
    a02 += rowp[-s2];
    a12 += rowp[kThreads - s2];
  }

  const int outStride = kB * kCout * kL;               // per-branch stride
  const int ob        = (b * kCout + co) * kL + t0;
  out[ob + tl]                         = a00;
  out[ob + tl + kThreads]              = a10;
  out[outStride + ob + tl]             = a01;
  out[outStride + ob + tl + kThreads]  = a11;
  out[2 * outStride + ob + tl]             = a02;
  out[2 * outStride + ob + tl + kThreads]  = a12;
}

extern "C" void kernel_launch(void* const* d_in, const int* in_sizes, int n_in,
                              void* d_out, int out_size, void* d_ws, size_t ws_size,
                              hipStream_t stream) {
  const float* x  = (const float*)d_in[0];
  const int*   i1 = (const int*)d_in[1];
  const int*   i2 = (const int*)d_in[2];
  const int*   i3 = (const int*)d_in[3];
  float*       o  = (float*)d_out;

  const int nblocks = kB * kCout * kNtiles;  // 4096
  addshift3_kernel<<<dim3(nblocks), dim3(kThreads), 0, stream>>>(x, i1, i2, i3, o);
}